// Warping_65094524339056
// MI455X (gfx1250) — compile-verified
//
#include <hip/hip_runtime.h>

// 3-D trilinear warp: out[b,x,y,z] = trilerp(image[b], (x,y,z) + ddf[b,x,y,z,:])
// Shapes fixed by the reference: B=4, fixed grid = moving image = 128^3, fp32.
//
// Memory-bound design for MI455X (23.3 TB/s HBM, 192 MB L2, wave32):
//  - ddf (100.7 MB) and out (33.5 MB) streamed with non-temporal hints so they
//    do not evict the L2-resident image (33.5 MB), which absorbs the 8-way
//    gather amplification.
//  - 4 consecutive z-voxels per thread => ddf read as three aligned b128 NT
//    loads per thread, output written as one aligned b128 NT store.
//  - All gathers/streams use uniform SGPR base + 32-bit VGPR element offset
//    (GVS addressing): one address VGPR per access instead of a 64-bit pair.
//  - grid-stride loop + global_prefetch_b8 of the next tile's ddf.

#define DIM   128
#define DIM2  (DIM * DIM)        // 16384
#define DIM3  (DIM * DIM * DIM)  // 2097152
#define BATCH 4
#define VPT   4                  // voxels per thread (contiguous in z)
#define TPB   256                // 8 waves per block (wave32)

typedef float v4f __attribute__((ext_vector_type(4)));

__device__ __forceinline__ float lerp1(float a, float b, float w) {
    return fmaf(w, b - a, a);   // a + w*(b-a)
}

__global__ __launch_bounds__(TPB) void warp_trilinear_kernel(
    const float* __restrict__ ddf,    // (B,128,128,128,3)
    const float* __restrict__ image,  // (B,128,128,128)
    float* __restrict__ out)          // (B,128,128,128)
{
    const unsigned nUnits       = BATCH * DIM3 / VPT;     // 2,097,152 thread-units
    const unsigned totalThreads = gridDim.x * blockDim.x;

    for (unsigned tu = blockIdx.x * blockDim.x + threadIdx.x;
         tu < nUnits; tu += totalThreads) {

        const unsigned vox0 = tu * VPT;                   // first voxel of this unit

        // Prefetch next iteration's ddf (each lane's byte pulls the whole
        // cacheline; covers the 48B/thread region) -> global_prefetch_b8.
        {
            const unsigned nextTu = tu + totalThreads;
            if (nextTu < nUnits) {
                __builtin_prefetch(ddf + (size_t)nextTu * (VPT * 3), 0, 0);
            }
        }

        // Decompose flat voxel index: vox = ((b*128 + x)*128 + y)*128 + z
        const int z0 = (int)(vox0 & (DIM - 1));
        const int y  = (int)((vox0 >> 7)  & (DIM - 1));
        const int x  = (int)((vox0 >> 14) & (DIM - 1));
        const unsigned bOff = (vox0 >> 21) * (unsigned)DIM3;  // batch term, folded into offsets

        // 12 displacement floats: three 16B-aligned non-temporal b128 loads,
        // 32-bit v4f-element offset against the uniform ddf base.
        const v4f* dbase = (const v4f*)ddf;
        const unsigned dq = tu * 3;                       // v4f units: vox0*3*4B / 16B
        const v4f d0 = __builtin_nontemporal_load(dbase + dq + 0);
        const v4f d1 = __builtin_nontemporal_load(dbase + dq + 1);
        const v4f d2 = __builtin_nontemporal_load(dbase + dq + 2);
        float dv[12];
        dv[0] = d0.x; dv[1]  = d0.y; dv[2]  = d0.z; dv[3]  = d0.w;
        dv[4] = d1.x; dv[5]  = d1.y; dv[6]  = d1.z; dv[7]  = d1.w;
        dv[8] = d2.x; dv[9]  = d2.y; dv[10] = d2.z; dv[11] = d2.w;

        v4f res;
        #pragma unroll
        for (int k = 0; k < VPT; ++k) {
            // Sample location in moving-image voxel coords, boundary-clamped.
            float lx = fminf(fmaxf((float)x        + dv[3 * k + 0], 0.f), (float)(DIM - 1));
            float ly = fminf(fmaxf((float)y        + dv[3 * k + 1], 0.f), (float)(DIM - 1));
            float lz = fminf(fmaxf((float)(z0 + k) + dv[3 * k + 2], 0.f), (float)(DIM - 1));

            const float fx = floorf(lx), fy = floorf(ly), fz = floorf(lz);
            const float wx = lx - fx,    wy = ly - fy,    wz = lz - fz;

            const int ix0 = (int)fx, iy0 = (int)fy, iz0 = (int)fz;
            const int ix1 = min(ix0 + 1, DIM - 1);
            const int iy1 = min(iy0 + 1, DIM - 1);
            const int iz1 = min(iz0 + 1, DIM - 1);

            // Flat 32-bit element offsets from the uniform image base
            // (GVS: saddr = image, voffset = 32-bit) — data-dependent gather,
            // serviced by the L2-resident image.
            const unsigned bx0 = bOff + (unsigned)(ix0 * DIM2);
            const unsigned bx1 = bOff + (unsigned)(ix1 * DIM2);
            const unsigned ry0 = (unsigned)(iy0 * DIM);
            const unsigned ry1 = (unsigned)(iy1 * DIM);
            const unsigned o00 = bx0 + ry0, o01 = bx0 + ry1;
            const unsigned o10 = bx1 + ry0, o11 = bx1 + ry1;

            const float v000 = image[o00 + (unsigned)iz0];
            const float v001 = image[o00 + (unsigned)iz1];
            const float v010 = image[o01 + (unsigned)iz0];
            const float v011 = image[o01 + (unsigned)iz1];
            const float v100 = image[o10 + (unsigned)iz0];
            const float v101 = image[o10 + (unsigned)iz1];
            const float v110 = image[o11 + (unsigned)iz0];
            const float v111 = image[o11 + (unsigned)iz1];

            // Lerp tree: 7 fused lerps == reference's 8-weight sum.
            const float c00 = lerp1(v000, v001, wz);
            const float c01 = lerp1(v010, v011, wz);
            const float c10 = lerp1(v100, v101, wz);
            const float c11 = lerp1(v110, v111, wz);
            const float c0  = lerp1(c00, c01, wy);
            const float c1  = lerp1(c10, c11, wy);
            res[k] = lerp1(c0, c1, wx);
        }

        // One aligned non-temporal b128 store of 4 contiguous outputs,
        // 32-bit v4f-element offset against the uniform out base.
        __builtin_nontemporal_store(res, (v4f*)out + tu);
    }
}

extern "C" void kernel_launch(void* const* d_in, const int* in_sizes, int n_in,
                              void* d_out, int out_size, void* d_ws, size_t ws_size,
                              hipStream_t stream) {
    (void)in_sizes; (void)n_in; (void)out_size; (void)d_ws; (void)ws_size;
    const float* ddf   = (const float*)d_in[0];  // (4,128,128,128,3) fp32
    const float* image = (const float*)d_in[1];  // (4,128,128,128)   fp32
    float*       out   = (float*)d_out;          // (4,128,128,128)   fp32

    // 2,097,152 thread-units / (2048*256) threads = 4 grid-stride iterations:
    // prefetch pipeline depth + plenty of waves to hide gather latency.
    const int blocks = 2048;
    warp_trilinear_kernel<<<blocks, TPB, 0, stream>>>(ddf, image, out);
}